// BrewCnnLayer_24240795419115
// MI455X (gfx1250) — compile-verified
//
#include <hip/hip_runtime.h>

typedef __attribute__((ext_vector_type(2))) float v2f;
typedef __attribute__((ext_vector_type(8))) float v8f;

// ---------------------------------------------------------------------------
// Problem constants
// ---------------------------------------------------------------------------
#define BATCH      2048
#define IN_HW      28
#define C1_K       7
#define C1_OHW     22                      // 28-7+1
#define C1_S       (C1_OHW*C1_OHW)         // 484
#define C1_KTOT    49
#define C1_KPAD    52                      // pad to multiple of 4
#define C2_K       5
#define C2_OHW     18                      // 22-5+1
#define C2_S       (C2_OHW*C2_OHW)         // 324
#define C2_KTOT    250                     // 10*5*5
#define C2_KPAD    252

#define N1         (BATCH*C1_S)            // 991232  (div by 32)
#define N2         (BATCH*C2_S)            // 663552  (div by 32)

#define H2_ELEMS   (BATCH*10*C2_S)         // 6,635,520
#define RAT1_ELEMS (BATCH*10*C1_S)         // 9,912,320
#define RAT1_OFF   H2_ELEMS
#define RAT2_OFF   (H2_ELEMS + RAT1_ELEMS) // 16,547,840

__device__ __forceinline__ void relu_ratio_store(v8f acc, int ocbase, int base,
                                                 int stride,
                                                 float* __restrict__ hdst,
                                                 float* __restrict__ rdst) {
#pragma unroll
    for (int j = 0; j < 8; ++j) {
        int oc = ocbase + j;
        if (oc < 10) {
            float hv = acc[j];
            float ho = hv > 0.0f ? hv : 0.0f;
            float rr = hv > 0.0f ? 1.0f : copysignf(0.0f, hv);
            hdst[base + oc * stride] = ho;   // GVS store: s[base] + vidx, scale_offset
            rdst[base + oc * stride] = rr;
        }
    }
}

// ---------------------------------------------------------------------------
// Layer 1: conv 1->10, 7x7.
// GEMM: M=16(pad of 10) x K=52(pad of 49) x N=32/wave (two 16-wide WMMA tiles)
// ---------------------------------------------------------------------------
__global__ __launch_bounds__(256) void conv1_wmma(
    const float* __restrict__ x,    // (2048, 784)  -- uniform SGPR base
    const float* __restrict__ W1,   // (10, 49)
    float*       __restrict__ h1,   // (2048,10,22,22) scratch
    float*       __restrict__ rat1) // (2048,10,22,22) output
{
    __shared__ float Wl[16 * C1_KPAD];   // zero-padded weights, row = oc
    __shared__ int   Dl[C1_KPAD];        // kidx -> input element-index delta

    const int tid = threadIdx.x;
    for (int i = tid; i < 16 * C1_KPAD; i += 256) {
        int oc = i / C1_KPAD, k = i - oc * C1_KPAD;
        Wl[i] = (oc < 10 && k < C1_KTOT) ? W1[oc * C1_KTOT + k] : 0.0f;
    }
    for (int i = tid; i < C1_KPAD; i += 256) {
        int kh = i / C1_K, kw = i - kh * C1_K;
        Dl[i] = (i < C1_KTOT) ? (kh * IN_HW + kw) : 0;  // padded K -> safe offset (weight is 0)
    }
    __syncthreads();

    const int lane  = tid & 31;
    const int wave  = tid >> 5;
    const int tile  = blockIdx.x * 8 + wave;     // 32 pixels per wave
    const int n     = lane & 15;

    // pixel 0 (N-tile 0) and pixel 1 (N-tile 1): keep 32-bit element indices,
    // let the hardware do saddr + vgpr_index*4 (GVS + scale_offset).
    const int p0 = tile * 32 + n;
    const int p1 = p0 + 16;
    const int b0 = p0 / C1_S, s0 = p0 - b0 * C1_S;
    const int b1 = p1 / C1_S, s1 = p1 - b1 * C1_S;
    const int oh0 = s0 / C1_OHW, ow0 = s0 - oh0 * C1_OHW;
    const int oh1 = s1 / C1_OHW, ow1 = s1 - oh1 * C1_OHW;
    const int q0 = b0 * (IN_HW * IN_HW) + oh0 * IN_HW + ow0;   // patch origin index
    const int q1 = b1 * (IN_HW * IN_HW) + oh1 * IN_HW + ow1;

    const int khalf = (lane >> 4) << 1;          // 0 for lanes 0-15, 2 for 16-31
    const v2f* wrow = (const v2f*)(&Wl[(lane & 15) * C1_KPAD]);

    v8f acc0 = {};
    v8f acc1 = {};
#pragma unroll
    for (int k = 0; k < C1_KPAD / 4; ++k) {      // 13 K-steps
        const int kb = 4 * k + khalf;
        v2f a = wrow[kb >> 1];                   // ds_load_b64, 8B aligned
        const int d0 = Dl[kb];
        const int d1 = Dl[kb + 1];
        v2f bb0, bb1;
        bb0.x = x[q0 + d0]; bb0.y = x[q0 + d1];  // GVS load: one v_add each
        bb1.x = x[q1 + d0]; bb1.y = x[q1 + d1];
        acc0 = __builtin_amdgcn_wmma_f32_16x16x4_f32(
            false, a, false, bb0, (short)0, acc0, false, false);
        acc1 = __builtin_amdgcn_wmma_f32_16x16x4_f32(
            false, a, false, bb1, (short)0, acc1, false, false);
    }

    // C/D layout: VGPR j -> (M=j, N=lane) lanes 0-15, (M=8+j, N=lane-16) lanes 16-31
    const int ocbase = (lane < 16) ? 0 : 8;
    relu_ratio_store(acc0, ocbase, b0 * (10 * C1_S) + s0, C1_S, h1, rat1);
    relu_ratio_store(acc1, ocbase, b1 * (10 * C1_S) + s1, C1_S, h1, rat1);
}

// ---------------------------------------------------------------------------
// Layer 2: conv 10->10, 5x5.
// GEMM: M=16(pad of 10) x K=252(pad of 250) x N=32/wave
// ---------------------------------------------------------------------------
__global__ __launch_bounds__(256) void conv2_wmma(
    const float* __restrict__ h1,   // (2048,10,22,22) -- uniform SGPR base
    const float* __restrict__ W2,   // (10, 250) in (ic,kh,kw) order
    float*       __restrict__ h2,   // (2048,10,18,18) output
    float*       __restrict__ rat2) // (2048,10,18,18) output
{
    __shared__ float Wl[16 * C2_KPAD];   // 16128 B
    __shared__ int   Dl[C2_KPAD];        // kidx -> h1 element-index delta

    const int tid = threadIdx.x;
    for (int i = tid; i < 16 * C2_KPAD; i += 256) {
        int oc = i / C2_KPAD, k = i - oc * C2_KPAD;
        Wl[i] = (oc < 10 && k < C2_KTOT) ? W2[oc * C2_KTOT + k] : 0.0f;
    }
    for (int i = tid; i < C2_KPAD; i += 256) {
        int ic = i / (C2_K * C2_K);
        int r  = i - ic * (C2_K * C2_K);
        int kh = r / C2_K, kw = r - kh * C2_K;
        Dl[i] = (i < C2_KTOT) ? (ic * C1_S + kh * C1_OHW + kw) : 0;
    }
    __syncthreads();

    const int lane  = tid & 31;
    const int wave  = tid >> 5;
    const int tile  = blockIdx.x * 8 + wave;     // 32 pixels per wave
    const int n     = lane & 15;

    const int p0 = tile * 32 + n;
    const int p1 = p0 + 16;
    const int b0 = p0 / C2_S, s0 = p0 - b0 * C2_S;
    const int b1 = p1 / C2_S, s1 = p1 - b1 * C2_S;
    const int oh0 = s0 / C2_OHW, ow0 = s0 - oh0 * C2_OHW;
    const int oh1 = s1 / C2_OHW, ow1 = s1 - oh1 * C2_OHW;
    const int q0 = b0 * (10 * C1_S) + oh0 * C1_OHW + ow0;   // patch origin index in h1
    const int q1 = b1 * (10 * C1_S) + oh1 * C1_OHW + ow1;

    const int khalf = (lane >> 4) << 1;
    const v2f* wrow = (const v2f*)(&Wl[(lane & 15) * C2_KPAD]);

    v8f acc0 = {};
    v8f acc1 = {};
#pragma unroll 7
    for (int k = 0; k < C2_KPAD / 4; ++k) {      // 63 K-steps, pipelined in chunks of 7
        const int kb = 4 * k + khalf;
        v2f a = wrow[kb >> 1];
        const int d0 = Dl[kb];
        const int d1 = Dl[kb + 1];
        v2f bb0, bb1;
        bb0.x = h1[q0 + d0]; bb0.y = h1[q0 + d1];
        bb1.x = h1[q1 + d0]; bb1.y = h1[q1 + d1];
        acc0 = __builtin_amdgcn_wmma_f32_16x16x4_f32(
            false, a, false, bb0, (short)0, acc0, false, false);
        acc1 = __builtin_amdgcn_wmma_f32_16x16x4_f32(
            false, a, false, bb1, (short)0, acc1, false, false);
    }

    const int ocbase = (lane < 16) ? 0 : 8;
    relu_ratio_store(acc0, ocbase, b0 * (10 * C2_S) + s0, C2_S, h2, rat2);
    relu_ratio_store(acc1, ocbase, b1 * (10 * C2_S) + s1, C2_S, h2, rat2);
}

// ---------------------------------------------------------------------------
// Launch
// ---------------------------------------------------------------------------
extern "C" void kernel_launch(void* const* d_in, const int* in_sizes, int n_in,
                              void* d_out, int out_size, void* d_ws, size_t ws_size,
                              hipStream_t stream) {
    const float* x  = (const float*)d_in[0];   // (2048, 784)
    const float* W1 = (const float*)d_in[1];   // (10, 49)
    const float* W2 = (const float*)d_in[2];   // (10, 250)

    float* out  = (float*)d_out;
    float* h2   = out;                 // h output, (2048,10,18,18)
    float* rat1 = out + RAT1_OFF;      // (2048,10,22,22)
    float* rat2 = out + RAT2_OFF;      // (2048,10,18,18)
    float* h1   = (float*)d_ws;        // intermediate, 39.65 MB (L2 resident)

    // N1/32 = 30976 wave-tiles, 8 waves per 256-thread block -> 3872 blocks
    conv1_wmma<<<N1 / 32 / 8, 256, 0, stream>>>(x, W1, h1, rat1);
    // N2/32 = 20736 wave-tiles -> 2592 blocks
    conv2_wmma<<<N2 / 32 / 8, 256, 0, stream>>>(h1, W2, h2, rat2);
}